// TrajectoryDecoder_58231166599657
// MI455X (gfx1250) — compile-verified
//
#include <hip/hip_runtime.h>
#include <hip/hip_bf16.h>

typedef __attribute__((ext_vector_type(2))) float v2f;
typedef __attribute__((ext_vector_type(8))) float v8f;

#define CAMS 6
#define TT   2
#define CF   256
#define FH   64
#define FW   160
#define NQ   128
#define EE   256
#define DFF  512
#define NH   8
#define HD   32
#define NP   150
#define NSAMP (TT*CAMS*NP)  // 1800
#define NLAYERS 6

// ---------------------------------------------------------------------------
// Generic WMMA GEMM: C[M x N] = act(scale * (A @ op(B)) + bias)
//   BT = true : C[m][n] = sum_k A[m*lda+k] * B[n*ldb+k]   (B is W[N][K], x @ W^T)
//   BT = false: C[m][n] = sum_k A[m*lda+k] * B[k*ldb+n]   (plain A @ B)
// One 16x16 output tile per wave, fp32 WMMA 16x16x4. blockDim = 256 (8 waves).
// gridDim.z batches (e.g. attention heads) via element strides azs/bzs/czs.
// ---------------------------------------------------------------------------
template<bool BT, bool RELU>
__global__ __launch_bounds__(256) void gemm_kernel(
    const float* __restrict__ A, int lda, int azs,
    const float* __restrict__ B, int ldb, int bzs,
    float* __restrict__ C, int ldc, int czs,
    const float* __restrict__ bias,
    int M, int N, int K, float scale)
{
  int z = blockIdx.z;
  A += (size_t)z * azs;
  B += (size_t)z * bzs;
  C += (size_t)z * czs;

  int ntn = N >> 4;
  int tile = blockIdx.x * 8 + (threadIdx.x >> 5);
  int total = (M >> 4) * ntn;
  if (tile >= total) return;
  int m0 = (tile / ntn) << 4;
  int n0 = (tile % ntn) << 4;

  int lane = threadIdx.x & 31;
  int half = lane >> 4;   // 0: K=0,1  1: K=2,3  (A layout); same split for B rows
  int l    = lane & 15;   // A row within tile / B row(NT) / C column

  const float* arow = A + (size_t)(m0 + l) * lda;
  const float* brow = BT ? (B + (size_t)(n0 + l) * ldb) : (B + (n0 + l));

  v8f c = {};
  for (int k = 0; k < K; k += 4) {
    int ka = k + half * 2;
    v2f a = *(const v2f*)(arow + ka);
    v2f b;
    if (BT) {
      b = *(const v2f*)(brow + ka);
    } else {
      b.x = brow[(size_t)ka * ldb];
      b.y = brow[(size_t)(ka + 1) * ldb];
    }
    // 8 args: (neg_a, A, neg_b, B, c_mod, C, reuse_a, reuse_b)
    c = __builtin_amdgcn_wmma_f32_16x16x4_f32(false, a, false, b, (short)0, c,
                                              false, false);
  }

  int nn = n0 + l;
  float bb = bias ? bias[nn] : 0.0f;
#pragma unroll
  for (int r = 0; r < 8; ++r) {
    int mm = m0 + r + half * 8;  // C layout: lanes 16-31 hold M=8..15
    float v = c[r] * scale + bb;
    if (RELU) v = fmaxf(v, 0.0f);
    C[(size_t)mm * ldc + nn] = v;
  }
}

// ---------------------------------------------------------------------------
// tp = hidden @ traj_w2^T + traj_b2  (N=11, too small for WMMA) -> tp buf + out
// ---------------------------------------------------------------------------
__global__ __launch_bounds__(256) void trajout_kernel(
    const float* __restrict__ hid, const float* __restrict__ w2,
    const float* __restrict__ b2, float* __restrict__ tp,
    float* __restrict__ out)
{
  int tid = blockIdx.x * blockDim.x + threadIdx.x;
  if (tid >= NQ * 11) return;
  int q = tid / 11, j = tid - q * 11;
  const float* h = hid + q * DFF;
  const float* w = w2 + j * DFF;
  float s = b2[j];
  for (int k = 0; k < DFF; ++k) s += h[k] * w[k];
  tp[tid] = s;
  out[tid] = s;
}

// ---------------------------------------------------------------------------
// Multi-camera deformable sampling. One block per query, 256 threads = channel.
// Phase 1: 1800 samples' geometry (transform+project+bilinear setup) into LDS.
// Phase 2: each channel thread gathers 4 taps per valid sample, accumulates.
// ---------------------------------------------------------------------------
__global__ __launch_bounds__(256) void sample_kernel(
    const float* __restrict__ feat, const float* __restrict__ calib,
    const float* __restrict__ ego, const float* __restrict__ tp,
    float* __restrict__ agg)
{
  __shared__ float s_wx[NSAMP], s_wy[NSAMP];
  __shared__ int s_x0[NSAMP], s_y0[NSAMP], s_boff[NSAMP], s_valid[NSAMP];

  int q = blockIdx.x;
  const float* t = tp + q * 11;

  for (int s = threadIdx.x; s < NSAMP; s += blockDim.x) {
    int tt = s / (CAMS * NP);
    int rem = s - tt * (CAMS * NP);
    int ci = rem / NP;
    int p = rem - ci * NP;

    // unit cube point
    int f = p / 25, ii = p - f * 25;
    float a = -1.0f + 0.5f * (float)(ii / 5);
    float b = -1.0f + 0.5f * (float)(ii % 5);
    float sg = (f & 1) ? 1.0f : -1.0f;
    int dim = f >> 1;
    float ux, uy, uz;
    if (dim == 0)      { ux = sg; uy = a;  uz = b; }
    else if (dim == 1) { ux = a;  uy = sg; uz = b; }
    else               { ux = a;  uy = b;  uz = sg; }

    // box local -> world (yaw + const-accel center)
    float yaw = t[7];
    float cyw = cosf(yaw), syw = sinf(yaw);
    float lx = ux * t[8], ly = uy * t[9], lz = uz * t[10];
    float rx = lx * cyw - ly * syw;
    float ry = lx * syw + ly * cyw;
    float dt = ego[tt * 4 + 3] - ego[(TT - 1) * 4 + 3];
    float cx = t[0] + t[3] * dt + 0.5f * t[5] * dt * dt;
    float cy2 = t[1] + t[4] * dt + 0.5f * t[6] * dt * dt;
    float wxp = rx + cx, wyp = ry + cy2, wzp = lz + t[2];

    // ego transform
    float ex = ego[tt * 4 + 0], ey = ego[tt * 4 + 1], eyaw = ego[tt * 4 + 2];
    float ce = cosf(eyaw), se = sinf(eyaw);
    float pxr = wxp - ex, pyr = wyp - ey;
    float gx = pxr * ce + pyr * se;
    float gy = -pxr * se + pyr * ce;

    // camera projection
    const float* cal = calib + ci * 8;
    float px = gx - cal[4], py = gy - cal[5], pz = wzp - cal[6];
    float cc = cosf(cal[7]), sc = sinf(cal[7]);
    float fwd = px * cc + py * sc;
    float left = -px * sc + py * cc;
    float zc = fwd, xc = -left, yc = -pz;
    float zs = fmaxf(zc, 0.1f);
    float u = cal[0] * xc / zs + cal[2];
    float v = cal[1] * yc / zs + cal[3];
    int valid = (zc > 0.1f) && (u >= 0.f) && (u < 1.f) && (v >= 0.f) && (v < 1.f);

    float pxs = u * (float)(FW - 1), pys = v * (float)(FH - 1);
    float x0f = floorf(pxs), y0f = floorf(pys);
    s_wx[s] = pxs - x0f;
    s_wy[s] = pys - y0f;
    int x0 = (int)x0f, y0 = (int)y0f;
    s_x0[s] = min(max(x0, 0), FW - 1);
    s_y0[s] = min(max(y0, 0), FH - 1);
    s_boff[s] = ((ci * TT + tt) * CF) * (FH * FW);
    s_valid[s] = valid;
  }
  __syncthreads();

  int c = threadIdx.x;
  const int plane = FH * FW;
  float acc = 0.0f;
  for (int s = 0; s < NSAMP; ++s) {
    if (!s_valid[s]) continue;
    const float* bse = feat + (size_t)(s_boff[s] + c * plane);
    int x0 = s_x0[s], y0 = s_y0[s];
    int x1 = min(x0 + 1, FW - 1), y1 = min(y0 + 1, FH - 1);
    float wx = s_wx[s], wy = s_wy[s];
    float f00 = bse[y0 * FW + x0];
    float f01 = bse[y0 * FW + x1];
    float f10 = bse[y1 * FW + x0];
    float f11 = bse[y1 * FW + x1];
    acc += f00 * (1.f - wx) * (1.f - wy) + f01 * wx * (1.f - wy)
         + f10 * (1.f - wx) * wy        + f11 * wx * wy;
  }
  agg[q * CF + c] = acc * (1.0f / (float)(NP * TT * CAMS));
}

// ---------------------------------------------------------------------------
// Row softmax over 128-wide attention rows. One block (128 thr) per row.
// ---------------------------------------------------------------------------
__global__ __launch_bounds__(128) void softmax_kernel(float* __restrict__ att)
{
  __shared__ float red[128];
  int row = blockIdx.x, tid = threadIdx.x;
  float x = att[row * NQ + tid];
  red[tid] = x;
  __syncthreads();
  for (int o = 64; o > 0; o >>= 1) {
    if (tid < o) red[tid] = fmaxf(red[tid], red[tid + o]);
    __syncthreads();
  }
  float mx = red[0];
  __syncthreads();
  float e = expf(x - mx);
  red[tid] = e;
  __syncthreads();
  for (int o = 64; o > 0; o >>= 1) {
    if (tid < o) red[tid] += red[tid + o];
    __syncthreads();
  }
  att[row * NQ + tid] = e / red[0];
}

// ---------------------------------------------------------------------------
// q = LayerNorm(q + res) * g + b. One block (256 thr) per row.
// ---------------------------------------------------------------------------
__global__ __launch_bounds__(256) void add_ln_kernel(
    float* __restrict__ q, const float* __restrict__ res,
    const float* __restrict__ g, const float* __restrict__ b)
{
  __shared__ float red[256];
  int row = blockIdx.x, tid = threadIdx.x;
  float x = q[row * EE + tid] + res[row * EE + tid];
  red[tid] = x;
  __syncthreads();
  for (int o = 128; o > 0; o >>= 1) {
    if (tid < o) red[tid] += red[tid + o];
    __syncthreads();
  }
  float mean = red[0] * (1.0f / EE);
  __syncthreads();
  float d = x - mean;
  red[tid] = d * d;
  __syncthreads();
  for (int o = 128; o > 0; o >>= 1) {
    if (tid < o) red[tid] += red[tid + o];
    __syncthreads();
  }
  float var = red[0] * (1.0f / EE);
  q[row * EE + tid] = d * rsqrtf(var + 1e-5f) * g[tid] + b[tid];
}

__global__ __launch_bounds__(256) void copy_kernel(
    float* __restrict__ dst, const float* __restrict__ src, int n)
{
  int i = blockIdx.x * blockDim.x + threadIdx.x;
  if (i < n) dst[i] = src[i];
}

// ---------------------------------------------------------------------------
// Host orchestration
// ---------------------------------------------------------------------------
static inline void gemm(hipStream_t st, bool bt, bool relu,
                        const float* A, int lda, int azs,
                        const float* B, int ldb, int bzs,
                        float* C, int ldc, int czs,
                        const float* bias, int M, int N, int K,
                        float scale, int Z)
{
  int tiles = (M / 16) * (N / 16);
  dim3 grid((tiles + 7) / 8, 1, Z);
  dim3 blk(256, 1, 1);
  if (bt) {
    if (relu)
      gemm_kernel<true, true><<<grid, blk, 0, st>>>(A, lda, azs, B, ldb, bzs,
                                                    C, ldc, czs, bias, M, N, K, scale);
    else
      gemm_kernel<true, false><<<grid, blk, 0, st>>>(A, lda, azs, B, ldb, bzs,
                                                     C, ldc, czs, bias, M, N, K, scale);
  } else {
    gemm_kernel<false, false><<<grid, blk, 0, st>>>(A, lda, azs, B, ldb, bzs,
                                                    C, ldc, czs, bias, M, N, K, scale);
  }
}

extern "C" void kernel_launch(void* const* d_in, const int* in_sizes, int n_in,
                              void* d_out, int out_size, void* d_ws, size_t ws_size,
                              hipStream_t stream) {
  const float* features = (const float*)d_in[0];
  const float* calib    = (const float*)d_in[1];
  const float* ego      = (const float*)d_in[2];
  const float* queries  = (const float*)d_in[3];
  // d_in[4] = query_pos (unused by reference forward)
  const float* traj_w1  = (const float*)d_in[5];
  const float* traj_b1  = (const float*)d_in[6];
  const float* traj_w2  = (const float*)d_in[7];
  const float* traj_b2  = (const float*)d_in[8];
  const float* fm_w1    = (const float*)d_in[9];
  const float* fm_b1    = (const float*)d_in[10];
  const float* fm_w2    = (const float*)d_in[11];
  const float* fm_b2    = (const float*)d_in[12];
  const float* sa_in_w  = (const float*)d_in[13];
  const float* sa_in_b  = (const float*)d_in[14];
  const float* sa_out_w = (const float*)d_in[15];
  const float* sa_out_b = (const float*)d_in[16];
  const float* ca_in_w  = (const float*)d_in[17];
  const float* ca_in_b  = (const float*)d_in[18];
  const float* ca_out_w = (const float*)d_in[19];
  const float* ca_out_b = (const float*)d_in[20];
  const float* l1_w     = (const float*)d_in[21];
  const float* l1_b     = (const float*)d_in[22];
  const float* l2_w     = (const float*)d_in[23];
  const float* l2_b     = (const float*)d_in[24];
  const float* n1_g     = (const float*)d_in[25];
  const float* n1_b     = (const float*)d_in[26];
  const float* n2_g     = (const float*)d_in[27];
  const float* n2_b     = (const float*)d_in[28];
  const float* n3_g     = (const float*)d_in[29];
  const float* n3_b     = (const float*)d_in[30];
  float* out = (float*)d_out;

  float* ws = (float*)d_ws;
  // workspace layout (float elements)
  float* Q    = ws;                     // 128*256
  float* HID  = Q    + NQ * EE;         // 128*512
  float* TP   = HID  + NQ * DFF;        // 128*11
  float* AGG  = TP   + NQ * 11;         // 128*256
  float* MEM  = AGG  + NQ * EE;         // 128*256
  float* QKV  = MEM  + NQ * EE;         // 128*768
  float* ATT  = QKV  + NQ * 3 * EE;     // 8*128*128
  float* AO   = ATT  + NH * NQ * NQ;    // 128*256
  float* PROJ = AO   + NQ * EE;         // 128*256

  const float inv_sqrt_d = 0.17677669529663689f;  // 1/sqrt(32)

  // q = queries (B=1 broadcast)
  copy_kernel<<<(NQ * EE + 255) / 256, 256, 0, stream>>>(Q, queries, NQ * EE);

  for (int li = 0; li < NLAYERS; ++li) {
    // ---- traj head: tp = relu(q@W1^T+b1)@W2^T+b2 ----
    gemm(stream, true, true, Q, EE, 0, traj_w1, EE, 0, HID, DFF, 0,
         traj_b1, NQ, DFF, EE, 1.0f, 1);
    trajout_kernel<<<(NQ * 11 + 255) / 256, 256, 0, stream>>>(
        HID, traj_w2, traj_b2, TP, out + li * NQ * 11);

    // ---- deformable multi-cam sampling -> agg ----
    sample_kernel<<<NQ, 256, 0, stream>>>(features, calib, ego, TP, AGG);

    // ---- feature MLP: mem = relu(agg@fm_w1^T+b)@fm_w2^T+b ----
    gemm(stream, true, true, AGG, EE, 0, fm_w1, EE, 0, HID, DFF, 0,
         fm_b1, NQ, DFF, EE, 1.0f, 1);
    gemm(stream, true, false, HID, DFF, 0, fm_w2, DFF, 0, MEM, EE, 0,
         fm_b2, NQ, EE, DFF, 1.0f, 1);

    // ---- self-attention ----
    gemm(stream, true, false, Q, EE, 0, sa_in_w + (size_t)li * 3 * EE * EE, EE, 0,
         QKV, 3 * EE, 0, sa_in_b + li * 3 * EE, NQ, 3 * EE, EE, 1.0f, 1);
    gemm(stream, true, false, QKV, 3 * EE, HD, QKV + EE, 3 * EE, HD,
         ATT, NQ, NQ * NQ, nullptr, NQ, NQ, HD, inv_sqrt_d, NH);
    softmax_kernel<<<NH * NQ, 128, 0, stream>>>(ATT);
    gemm(stream, false, false, ATT, NQ, NQ * NQ, QKV + 2 * EE, 3 * EE, HD,
         AO, EE, HD, nullptr, NQ, HD, NQ, 1.0f, NH);
    gemm(stream, true, false, AO, EE, 0, sa_out_w + (size_t)li * EE * EE, EE, 0,
         PROJ, EE, 0, sa_out_b + li * EE, NQ, EE, EE, 1.0f, 1);
    add_ln_kernel<<<NQ, 256, 0, stream>>>(Q, PROJ, n1_g + li * EE, n1_b + li * EE);

    // ---- cross-attention (k,v from mem) ----
    const float* caw = ca_in_w + (size_t)li * 3 * EE * EE;
    const float* cab = ca_in_b + li * 3 * EE;
    gemm(stream, true, false, Q, EE, 0, caw, EE, 0, QKV, 3 * EE, 0,
         cab, NQ, EE, EE, 1.0f, 1);
    gemm(stream, true, false, MEM, EE, 0, caw + EE * EE, EE, 0, QKV + EE, 3 * EE, 0,
         cab + EE, NQ, EE, EE, 1.0f, 1);
    gemm(stream, true, false, MEM, EE, 0, caw + 2 * EE * EE, EE, 0, QKV + 2 * EE, 3 * EE, 0,
         cab + 2 * EE, NQ, EE, EE, 1.0f, 1);
    gemm(stream, true, false, QKV, 3 * EE, HD, QKV + EE, 3 * EE, HD,
         ATT, NQ, NQ * NQ, nullptr, NQ, NQ, HD, inv_sqrt_d, NH);
    softmax_kernel<<<NH * NQ, 128, 0, stream>>>(ATT);
    gemm(stream, false, false, ATT, NQ, NQ * NQ, QKV + 2 * EE, 3 * EE, HD,
         AO, EE, HD, nullptr, NQ, HD, NQ, 1.0f, NH);
    gemm(stream, true, false, AO, EE, 0, ca_out_w + (size_t)li * EE * EE, EE, 0,
         PROJ, EE, 0, ca_out_b + li * EE, NQ, EE, EE, 1.0f, 1);
    add_ln_kernel<<<NQ, 256, 0, stream>>>(Q, PROJ, n2_g + li * EE, n2_b + li * EE);

    // ---- FFN ----
    gemm(stream, true, true, Q, EE, 0, l1_w + (size_t)li * DFF * EE, EE, 0,
         HID, DFF, 0, l1_b + li * DFF, NQ, DFF, EE, 1.0f, 1);
    gemm(stream, true, false, HID, DFF, 0, l2_w + (size_t)li * EE * DFF, DFF, 0,
         PROJ, EE, 0, l2_b + li * EE, NQ, EE, DFF, 1.0f, 1);
    add_ln_kernel<<<NQ, 256, 0, stream>>>(Q, PROJ, n3_g + li * EE, n3_b + li * EE);
  }

  // ---- final traj head ----
  gemm(stream, true, true, Q, EE, 0, traj_w1, EE, 0, HID, DFF, 0,
       traj_b1, NQ, DFF, EE, 1.0f, 1);
  trajout_kernel<<<(NQ * 11 + 255) / 256, 256, 0, stream>>>(
      HID, traj_w2, traj_b2, TP, out + NLAYERS * NQ * 11);
}